// TupletLoss_19765439496304
// MI455X (gfx1250) — compile-verified
//
#include <hip/hip_runtime.h>

typedef float v2f __attribute__((ext_vector_type(2)));
typedef float v8f __attribute__((ext_vector_type(8)));

#define NPTS 8192
#define DIM  128
#define NT   (NPTS / 16)   // 512 column tiles of 16
#define MARGIN 0.0f

// ---------------------------------------------------------------------------
// Kernel 1: per-row squared norms. One wave32 per row, float4 loads.
// ---------------------------------------------------------------------------
__global__ __launch_bounds__(256) void row_sq_kernel(const float* __restrict__ X,
                                                     float* __restrict__ sq) {
  int lane = threadIdx.x & 31;
  int w    = threadIdx.x >> 5;
  int row  = blockIdx.x * 8 + w;
  const float4* p = (const float4*)(X + (size_t)row * DIM) + lane;  // 32 lanes * 4 = 128
  float4 v = *p;
  float s = v.x * v.x + v.y * v.y + v.z * v.z + v.w * v.w;
  s += __shfl_xor(s, 16, 32);
  s += __shfl_xor(s, 8, 32);
  s += __shfl_xor(s, 4, 32);
  s += __shfl_xor(s, 2, 32);
  s += __shfl_xor(s, 1, 32);
  if (lane == 0) sq[row] = s;
}

// ---------------------------------------------------------------------------
// Kernel 2: fused Gram (WMMA fp32) + hardest-positive / hardest-negative.
// Block = 8 waves = 256 threads, owns a 16-row tile. Each wave scans 1/8 of
// the 512 column tiles. A-tile (16x128) is resident in 64 VGPRs per wave.
//
// The K=128 reduction is split into FOUR independent WMMA accumulator chains
// (depth 8 each) so the scheduler can overlap WMMA latency instead of
// serializing 32 dependent v_wmma ops; partial C tiles are summed at the end.
//
// fp32 16x16x4 WMMA operand layout (ISA 7.12.2):
//   A (16x4):  lanes 0-15 -> M=lane, VGPR{0,1}=K{0,1}; lanes 16-31 -> K{2,3}
//   B (4x16):  lanes 0-15 -> N=lane, VGPR{0,1}=K{0,1}; lanes 16-31 -> K{2,3}
//              (B[k][n] = X[n][k] for Gram, so same per-lane load pattern)
//   C/D:       VGPR r, lanes 0-15 -> D[M=r][N=lane]; lanes 16-31 -> D[M=r+8][N=lane-16]
// ---------------------------------------------------------------------------
__global__ __launch_bounds__(256) void gram_hardest_kernel(
    const float* __restrict__ X, const int* __restrict__ tgt,
    const float* __restrict__ sq, float* __restrict__ dist_ap,
    float* __restrict__ dist_an) {
  __shared__ float red_max[8][16];
  __shared__ float red_min[8][16];

  int lane = threadIdx.x & 31;
  int w    = threadIdx.x >> 5;
  int hi   = lane >> 4;     // 0: K pair {0,1}/M base 0 ; 1: K pair {2,3}/M base 8
  int lo   = lane & 15;
  int rowBase = blockIdx.x * 16;

  // Resident A tile: a[s] = X[rowBase+lo][4s + 2*hi .. +1]
  int koff = hi * 2;
  const float* Xa = X + (size_t)(rowBase + lo) * DIM + koff;
  v2f a[32];
#pragma unroll
  for (int s = 0; s < 32; ++s) a[s] = *(const v2f*)(Xa + 4 * s);

  // Per-lane row-side constants (fixed M per C-VGPR-row r)
  float sqm[8];
  int   tm[8];
#pragma unroll
  for (int r = 0; r < 8; ++r) {
    int m = rowBase + r + 8 * hi;
    sqm[r] = sq[m];
    tm[r]  = tgt[m];
  }

  float mp[8], mn[8];
#pragma unroll
  for (int r = 0; r < 8; ++r) {
    mp[r] = -__builtin_inff();
    mn[r] =  __builtin_inff();
  }

  for (int j = w; j < NT; j += 8) {
    int colBase = j * 16;
    const float* Xb = X + (size_t)(colBase + lo) * DIM + koff;
    float sqn = sq[colBase + lo];
    int   tn  = tgt[colBase + lo];

    v2f b[32];
#pragma unroll
    for (int s = 0; s < 32; ++s) b[s] = *(const v2f*)(Xb + 4 * s);

    // 4 independent accumulator chains, interleaved (ILP = 4)
    v8f c0 = {0.f, 0.f, 0.f, 0.f, 0.f, 0.f, 0.f, 0.f};
    v8f c1 = c0, c2 = c0, c3 = c0;
#pragma unroll
    for (int s = 0; s < 8; ++s) {
      c0 = __builtin_amdgcn_wmma_f32_16x16x4_f32(false, a[4 * s + 0], false,
                                                 b[4 * s + 0], (short)0, c0,
                                                 false, false);
      c1 = __builtin_amdgcn_wmma_f32_16x16x4_f32(false, a[4 * s + 1], false,
                                                 b[4 * s + 1], (short)0, c1,
                                                 false, false);
      c2 = __builtin_amdgcn_wmma_f32_16x16x4_f32(false, a[4 * s + 2], false,
                                                 b[4 * s + 2], (short)0, c2,
                                                 false, false);
      c3 = __builtin_amdgcn_wmma_f32_16x16x4_f32(false, a[4 * s + 3], false,
                                                 b[4 * s + 3], (short)0, c3,
                                                 false, false);
    }

#pragma unroll
    for (int r = 0; r < 8; ++r) {
      float dot  = (c0[r] + c1[r]) + (c2[r] + c3[r]);
      float d2   = sqm[r] + sqn - 2.0f * dot;
      float dist = sqrtf(fmaxf(d2, 1e-12f));
      bool  same = (tm[r] == tn);
      // branchless -> v_cndmask, EXEC stays all-ones around WMMA
      mp[r] = same ? fmaxf(mp[r], dist) : mp[r];
      mn[r] = same ? mn[r] : fminf(mn[r], dist);
    }
  }

  // Reduce across the 16 lanes of each half-wave (same M, different N)
#pragma unroll
  for (int r = 0; r < 8; ++r) {
#pragma unroll
    for (int off = 8; off >= 1; off >>= 1) {
      mp[r] = fmaxf(mp[r], __shfl_xor(mp[r], off, 32));
      mn[r] = fminf(mn[r], __shfl_xor(mn[r], off, 32));
    }
    if (lo == 0) {
      red_max[w][r + 8 * hi] = mp[r];
      red_min[w][r + 8 * hi] = mn[r];
    }
  }
  __syncthreads();

  // Combine the 8 waves' partials; 16 threads write the 16 rows of this tile
  if (threadIdx.x < 16) {
    float ap = red_max[0][threadIdx.x];
    float an = red_min[0][threadIdx.x];
#pragma unroll
    for (int ww = 1; ww < 8; ++ww) {
      ap = fmaxf(ap, red_max[ww][threadIdx.x]);
      an = fminf(an, red_min[ww][threadIdx.x]);
    }
    dist_ap[rowBase + threadIdx.x] = ap;
    dist_an[rowBase + threadIdx.x] = an;
  }
}

// ---------------------------------------------------------------------------
// Kernel 3: per-block partial sums of loss and precision terms (deterministic)
// ---------------------------------------------------------------------------
__global__ __launch_bounds__(256) void pairs_partial_kernel(
    const float* __restrict__ dist_ap, const float* __restrict__ dist_an,
    float* __restrict__ pl, float* __restrict__ pp) {
  __shared__ float sl[256];
  __shared__ float sp[256];
  int idx = blockIdx.x * 256 + threadIdx.x;
  float pa = __expf(-dist_ap[idx]);   // y = +1 element
  float pn = __expf(-dist_an[idx]);   // y = -1 element
  float loss = pa + fmaxf(MARGIN - pn, 0.0f);
  float c1 = ((pa < MARGIN) == true)  ? 1.0f : 0.0f;
  float c2 = ((pn < MARGIN) == false) ? 1.0f : 0.0f;
  sl[threadIdx.x] = loss;
  sp[threadIdx.x] = c1 + c2;
  __syncthreads();
  for (int s = 128; s > 0; s >>= 1) {
    if (threadIdx.x < s) {
      sl[threadIdx.x] += sl[threadIdx.x + s];
      sp[threadIdx.x] += sp[threadIdx.x + s];
    }
    __syncthreads();
  }
  if (threadIdx.x == 0) {
    pl[blockIdx.x] = sl[0];
    pp[blockIdx.x] = sp[0];
  }
}

__global__ void finalize_kernel(const float* __restrict__ pl,
                                const float* __restrict__ pp,
                                float* __restrict__ out) {
  int lane = threadIdx.x;  // 32 threads, 32 partials
  float l = pl[lane];
  float p = pp[lane];
  for (int off = 16; off >= 1; off >>= 1) {
    l += __shfl_xor(l, off, 32);
    p += __shfl_xor(p, off, 32);
  }
  if (lane == 0) {
    out[0] = l / (2.0f * NPTS);
    out[1] = p / (2.0f * NPTS);
  }
}

// ---------------------------------------------------------------------------
extern "C" void kernel_launch(void* const* d_in, const int* in_sizes, int n_in,
                              void* d_out, int out_size, void* d_ws, size_t ws_size,
                              hipStream_t stream) {
  const float* X   = (const float*)d_in[0];
  const int*   tgt = (const int*)d_in[1];

  float* ws      = (float*)d_ws;
  float* sq      = ws;                 // [8192]
  float* dist_ap = ws + NPTS;          // [8192]
  float* dist_an = ws + 2 * NPTS;      // [8192]
  float* pl      = ws + 3 * NPTS;      // [32]
  float* pp      = ws + 3 * NPTS + 32; // [32]

  row_sq_kernel<<<NPTS / 8, 256, 0, stream>>>(X, sq);
  gram_hardest_kernel<<<NPTS / 16, 256, 0, stream>>>(X, tgt, sq, dist_ap, dist_an);
  pairs_partial_kernel<<<NPTS / 256, 256, 0, stream>>>(dist_ap, dist_an, pl, pp);
  finalize_kernel<<<1, 32, 0, stream>>>(pl, pp, (float*)d_out);
}